// FilteredNoiseGenerator_79731772883058
// MI455X (gfx1250) — compile-verified
//
#include <hip/hip_runtime.h>

typedef __attribute__((ext_vector_type(2))) float v2f;
typedef __attribute__((ext_vector_type(8))) float v8f;

#define FPI 3.14159265358979323846f

static constexpr int  B        = 4;
static constexpr int  NFRAMES  = 500;
static constexpr int  NBANKS   = 64;
static constexpr int  NSAMP    = 256000;   // 500 * 512
static constexpr int  N1       = 500;      // padded to 512
static constexpr int  N2       = 512;
static constexpr int  MDIM     = 512;      // all GEMMs are 512^3 complex
static constexpr long long MSZ = 512LL * 512LL;
static constexpr int  NFREQ    = 128001;   // rfft bins
static constexpr int  HP       = 128016;   // padded per-batch H stride

// ---------------- workspace layout (float offsets) ----------------
static constexpr long long OFF_F500R = 0;
static constexpr long long OFF_F500I = OFF_F500R + MSZ;
static constexpr long long OFF_F512R = OFF_F500I + MSZ;
static constexpr long long OFF_F512I = OFF_F512R + MSZ;
static constexpr long long OFF_TWR   = OFF_F512I + MSZ;
static constexpr long long OFF_TWI   = OFF_TWR + MSZ;
static constexpr long long OFF_H     = OFF_TWI + MSZ;           // B * HP
static constexpr long long OFF_W     = OFF_H + (long long)B * HP; // 512
static constexpr long long OFF_MM    = OFF_W + 512;             // B*64
static constexpr long long OFF_P0R   = OFF_MM + 256;
static constexpr long long OFF_P0I   = OFF_P0R + (long long)B * MSZ;
static constexpr long long OFF_P1R   = OFF_P0I + (long long)B * MSZ;
static constexpr long long OFF_P1I   = OFF_P1R + (long long)B * MSZ;

// ---------------- constant matrices: F500, F512, twiddle ----------------
__global__ __launch_bounds__(256)
void build_consts(float* __restrict__ f500r, float* __restrict__ f500i,
                  float* __restrict__ f512r, float* __restrict__ f512i,
                  float* __restrict__ twr,   float* __restrict__ twi) {
    int idx = blockIdx.x * blockDim.x + threadIdx.x;
    if (idx >= 512 * 512) return;
    int i = idx >> 9, j = idx & 511;

    // F512[i][j] = exp(-2*pi*I * i*j / 512)
    {
        int m = (i * j) & 511;
        float th = (float)m * (2.0f * FPI / 512.0f);
        f512r[idx] =  cosf(th);
        f512i[idx] = -sinf(th);
    }
    // F500[i][j] = exp(-2*pi*I * i*j / 500), zero-padded to 512x512
    if (i < N1 && j < N1) {
        int m = (i * j) % N1;
        float th = (float)m * (2.0f * FPI / (float)N1);
        f500r[idx] =  cosf(th);
        f500i[idx] = -sinf(th);
    } else {
        f500r[idx] = 0.0f;
        f500i[idx] = 0.0f;
    }
    // forward twiddle Tw[k1][n2] = exp(-2*pi*I * k1*n2 / 256000)
    {
        int m = (i * j) % NSAMP;
        float th = (float)m * (2.0f * FPI / (float)NSAMP);
        twr[idx] =  cosf(th);
        twi[idx] = -sinf(th);
    }
}

// ---------------- frame-mean weights (exact, gather, no atomics) ------
__global__ __launch_bounds__(256)
void mean_weights(float* __restrict__ w) {
    int f = blockIdx.x * blockDim.x + threadIdx.x;
    if (f >= 512) return;
    if (f >= NFRAMES) { w[f] = 0.0f; return; }
    const float step = (float)(NFRAMES - 1) / (float)(NSAMP - 1);
    const float inv_step = (float)(NSAMP - 1) / (float)(NFRAMES - 1);
    int j0 = (int)((float)(f - 1) * inv_step) - 2; if (j0 < 0) j0 = 0;
    int j1 = (int)((float)(f + 1) * inv_step) + 2; if (j1 > NSAMP - 1) j1 = NSAMP - 1;
    float acc = 0.0f;
    for (int j = j0; j <= j1; ++j) {
        float pos = (float)j * step;
        int lo = (int)pos;
        if (lo > NFRAMES - 2) lo = NFRAMES - 2;
        float frac = pos - (float)lo;
        if (lo == f)     acc += 1.0f - frac;
        if (lo + 1 == f) acc += frac;
    }
    w[f] = acc / (float)NSAMP;
}

// ---------------- mean_mag[b][j] = sum_f fm[b][f][j] * w[f] -----------
__global__ __launch_bounds__(256)
void mean_mag_k(const float* __restrict__ fm, const float* __restrict__ w,
                float* __restrict__ mm) {
    int t = blockIdx.x * blockDim.x + threadIdx.x;
    if (t >= B * NBANKS) return;
    int b = t / NBANKS, j = t % NBANKS;
    const float* p = fm + (long long)b * NFRAMES * NBANKS + j;
    float acc = 0.0f;
    for (int f = 0; f < NFRAMES; ++f) acc += p[(long long)f * NBANKS] * w[f];
    mm[t] = acc;
}

// ---------------- filter response H[b][k] over rfft bins --------------
__global__ __launch_bounds__(256)
void response_k(const float* __restrict__ mm, float* __restrict__ H) {
    int k = blockIdx.x * blockDim.x + threadIdx.x;
    if (k >= NFREQ) return;
    float freq = (float)k * (22050.0f / (float)NSAMP);
    float lx = logf(freq + 1e-7f);
    const float l0 = logf(20.0f);
    const float dl = (logf(11025.0f) - l0) / 63.0f;
    float sum = 0.0f, a0 = 0.0f, a1 = 0.0f, a2 = 0.0f, a3 = 0.0f;
    for (int j = 0; j < NBANKS; ++j) {
        float ff = expf(l0 + (float)j * dl);
        float d  = lx - logf(ff + 1e-7f);
        float wj = expf(-2.0f * d * d);   // exp(-d^2 / 0.5)
        sum += wj;
        a0 += mm[j]            * wj;
        a1 += mm[NBANKS + j]   * wj;
        a2 += mm[2*NBANKS + j] * wj;
        a3 += mm[3*NBANKS + j] * wj;
    }
    float inv = 1.0f / (sum + 1e-7f);
    H[k]          = a0 * inv;
    H[HP + k]     = a1 * inv;
    H[2*HP + k]   = a2 * inv;
    H[3*HP + k]   = a3 * inv;
}

// ---------------- pack noise -> padded 512x512 real matrix ------------
__global__ __launch_bounds__(256)
void pack_k(const float* __restrict__ noise, float* __restrict__ pr,
            float* __restrict__ pi) {
    int idx = blockIdx.x * blockDim.x + threadIdx.x;
    if (idx >= B * 512 * 512) return;
    int b = idx >> 18, rem = idx & 262143;
    int n1 = rem >> 9, n2 = rem & 511;
    pr[idx] = (n1 < N1) ? noise[(long long)b * NSAMP + n1 * N2 + n2] : 0.0f;
    pi[idx] = 0.0f;
}

// ---------------- complex GEMM 512x512x512 via f32 WMMA ---------------
// C = A' * B' with A' = (Ar, sa*Ai), B' = (Br, sb*Bi)  (sa,sb = +/-1 for conj)
__global__ __launch_bounds__(128)
void cgemm512(const float* __restrict__ Ar, const float* __restrict__ Ai, long long sA,
              const float* __restrict__ Br, const float* __restrict__ Bi, long long sB,
              float* __restrict__ Cr, float* __restrict__ Ci, long long sC,
              float sa, float sb) {
    const long long bo = (long long)blockIdx.z;
    Ar += bo * sA; Ai += bo * sA;
    Br += bo * sB; Bi += bo * sB;
    Cr += bo * sC; Ci += bo * sC;

    const int lane = threadIdx.x;
    const int lo = lane & 15, hi = lane >> 4;
    const int tileM = (blockIdx.y * 4 + threadIdx.y) * 16;
    const int tileN = blockIdx.x * 16;

    v8f prr = {}, pri = {}, pir = {}, pii = {};
    const int rowA = tileM + lo;
    const int colB = tileN + lo;

    for (int k0 = 0; k0 < MDIM; k0 += 4) {
        const int ka = k0 + 2 * hi;
        v2f ar = *(const v2f*)(Ar + (long long)rowA * MDIM + ka);
        v2f ai = *(const v2f*)(Ai + (long long)rowA * MDIM + ka);
        v2f br, bi;
        br[0] = Br[(long long)ka * MDIM + colB];
        br[1] = Br[(long long)(ka + 1) * MDIM + colB];
        bi[0] = Bi[(long long)ka * MDIM + colB];
        bi[1] = Bi[(long long)(ka + 1) * MDIM + colB];

        prr = __builtin_amdgcn_wmma_f32_16x16x4_f32(false, ar, false, br, (short)0, prr, false, false);
        pii = __builtin_amdgcn_wmma_f32_16x16x4_f32(false, ai, false, bi, (short)0, pii, false, false);
        pir = __builtin_amdgcn_wmma_f32_16x16x4_f32(false, ai, false, br, (short)0, pir, false, false);
        pri = __builtin_amdgcn_wmma_f32_16x16x4_f32(false, ar, false, bi, (short)0, pri, false, false);
    }

    const float ss = sa * sb;
    for (int j = 0; j < 8; ++j) {
        int row = tileM + j + 8 * hi;
        long long idx = (long long)row * MDIM + tileN + lo;
        Cr[idx] = prr[j] - ss * pii[j];
        Ci[idx] = sa * pir[j] + sb * pri[j];
    }
}

// ---------------- in-place twiddle multiply (s=+1 fwd, -1 inv) --------
__global__ __launch_bounds__(256)
void twiddle_k(float* __restrict__ pr, float* __restrict__ pi,
               const float* __restrict__ twr, const float* __restrict__ twi,
               float s) {
    int idx = blockIdx.x * blockDim.x + threadIdx.x;
    if (idx >= B * 512 * 512) return;
    int rem = idx & 262143;
    float tr = twr[rem];
    float ti = s * twi[rem];
    float cr = pr[idx], ci = pi[idx];
    pr[idx] = cr * tr - ci * ti;
    pi[idx] = cr * ti + ci * tr;
}

// ---------------- multiply spectrum by H (mirror upper half) ----------
__global__ __launch_bounds__(256)
void applyH_k(float* __restrict__ pr, float* __restrict__ pi,
              const float* __restrict__ H) {
    int idx = blockIdx.x * blockDim.x + threadIdx.x;
    if (idx >= B * 512 * 512) return;
    int b = idx >> 18, rem = idx & 262143;
    int k1 = rem >> 9, k2 = rem & 511;
    float h = 0.0f;
    if (k1 < N1) {
        int k = N1 * k2 + k1;                 // full-spectrum index, < 256000
        int r = (k <= NSAMP / 2) ? k : NSAMP - k;
        h = H[(long long)b * HP + r];
    }
    pr[idx] *= h;
    pi[idx] *= h;
}

// ---------------- unpack real part, scale 1/N -------------------------
__global__ __launch_bounds__(256)
void unpack_k(const float* __restrict__ pr, float* __restrict__ out) {
    int idx = blockIdx.x * blockDim.x + threadIdx.x;
    if (idx >= B * NSAMP) return;
    int b = idx / NSAMP, n = idx % NSAMP;
    int n1 = n >> 9, n2 = n & 511;
    out[idx] = pr[(long long)b * MSZ + (long long)n1 * MDIM + n2] * (1.0f / (float)NSAMP);
}

extern "C" void kernel_launch(void* const* d_in, const int* in_sizes, int n_in,
                              void* d_out, int out_size, void* d_ws, size_t ws_size,
                              hipStream_t stream) {
    const float* fm    = (const float*)d_in[0];   // (4, 500, 64) f32
    const float* noise = (const float*)d_in[1];   // (4, 256000) f32
    float* out = (float*)d_out;
    float* ws  = (float*)d_ws;

    float* f500r = ws + OFF_F500R;  float* f500i = ws + OFF_F500I;
    float* f512r = ws + OFF_F512R;  float* f512i = ws + OFF_F512I;
    float* twr   = ws + OFF_TWR;    float* twi   = ws + OFF_TWI;
    float* H     = ws + OFF_H;
    float* w     = ws + OFF_W;
    float* mm    = ws + OFF_MM;
    float* p0r   = ws + OFF_P0R;    float* p0i   = ws + OFF_P0I;
    float* p1r   = ws + OFF_P1R;    float* p1i   = ws + OFF_P1I;

    const int T = 256;
    const int nEl  = B * 512 * 512;

    build_consts<<<(512 * 512 + T - 1) / T, T, 0, stream>>>(f500r, f500i, f512r, f512i, twr, twi);
    mean_weights<<<2, T, 0, stream>>>(w);
    mean_mag_k<<<1, T, 0, stream>>>(fm, w, mm);
    response_k<<<(NFREQ + T - 1) / T, T, 0, stream>>>(mm, H);
    pack_k<<<(nEl + T - 1) / T, T, 0, stream>>>(noise, p0r, p0i);

    dim3 gg(32, 8, B), bb(32, 4, 1);
    // forward: C1 = F500 * A
    cgemm512<<<gg, bb, 0, stream>>>(f500r, f500i, 0, p0r, p0i, MSZ, p1r, p1i, MSZ, 1.0f, 1.0f);
    twiddle_k<<<(nEl + T - 1) / T, T, 0, stream>>>(p1r, p1i, twr, twi, 1.0f);
    // D = C1 * F512   (full forward spectrum, D[k1][k2] = X[500*k2 + k1])
    cgemm512<<<gg, bb, 0, stream>>>(p1r, p1i, MSZ, f512r, f512i, 0, p0r, p0i, MSZ, 1.0f, 1.0f);
    applyH_k<<<(nEl + T - 1) / T, T, 0, stream>>>(p0r, p0i, H);
    // inverse: E = D' * conj(F512)
    cgemm512<<<gg, bb, 0, stream>>>(p0r, p0i, MSZ, f512r, f512i, 0, p1r, p1i, MSZ, 1.0f, -1.0f);
    twiddle_k<<<(nEl + T - 1) / T, T, 0, stream>>>(p1r, p1i, twr, twi, -1.0f);
    // out_mat = conj(F500) * E'
    cgemm512<<<gg, bb, 0, stream>>>(f500r, f500i, 0, p1r, p1i, MSZ, p0r, p0i, MSZ, -1.0f, 1.0f);
    unpack_k<<<(B * NSAMP + T - 1) / T, T, 0, stream>>>(p0r, out);
}